// SpectralConv1D_75282186764378
// MI455X (gfx1250) — compile-verified
//
#include <hip/hip_runtime.h>

// ---------------------------------------------------------------------------
// SpectralConv1D as three fp32 WMMA GEMMs on gfx1250 (wave32).
//   B=64, S=4096, Cin=Cout=64, MODES=32  -> R = 2*MODES = 64 real DFT rows.
//
// Stage 0: build Wf[64][4096] (cos / -sin rows) and Wi[4096][64] (irfft
//          weights with 1/S & 2/S scaling folded in), phases reduced mod S.
// Stage 1: Xp[s][b] (64x64) = Wf[:, sK..] @ x[b][sK.., :]   (K split 4 ways,
//          256 blocks; x staged through LDS with b128 loads)
// Stage 2: F[b,2m+ri,:] = (sum_s Xp[s][b,2m+ri,:]) @ K[m]   (reduction fused)
// Stage 3: out[b] (4096x64) = Wi (4096x64) @ F[b] (64x64)
// ---------------------------------------------------------------------------

typedef __attribute__((ext_vector_type(2))) float v2f;
typedef __attribute__((ext_vector_type(8))) float v8f;

constexpr int kS     = 4096;
constexpr int kB     = 64;
constexpr int kCin   = 64;
constexpr int kCout  = 64;
constexpr int kModes = 32;
constexpr int kR     = 2 * kModes;        // 64 real rows (re/im interleaved)
constexpr int kSplit = 4;                 // K-split factor for stage 1
constexpr int kKPer  = kS / kSplit;       // 1024 K per block
constexpr int kChunk = 64;                // k-rows staged in LDS per step
constexpr int kPad   = 72;                // padded LDS row stride (floats)
constexpr int kXP    = kB * kR * kCin;    // 262144 floats per partial

__device__ __forceinline__ v8f wmma_f32(v2f a, v2f b, v8f c) {
  // D(16x16,f32) = A(16x4,f32) * B(4x16,f32) + C
  return __builtin_amdgcn_wmma_f32_16x16x4_f32(
      /*neg_a=*/false, a, /*neg_b=*/false, b,
      /*c_mod=*/(short)0, c, /*reuse_a=*/false, /*reuse_b=*/false);
}

// ---------------------------------------------------------------------------
// Stage 0: twiddle tables. 262144 threads; each fills one element of Wf and
// one of Wi. Phase index reduced exactly mod S (power of two) before sincos.
// ---------------------------------------------------------------------------
__global__ void twiddle_kernel(float* __restrict__ Wf, float* __restrict__ Wi) {
  const int tid = blockIdx.x * blockDim.x + threadIdx.x;      // 0 .. 262143
  const float two_pi_over_s = 6.283185307179586476925f / (float)kS;
  {
    // Wf[r][s]: r = 2m -> cos(2*pi*m*s/S), r = 2m+1 -> -sin(2*pi*m*s/S)
    const int r = tid >> 12, s = tid & (kS - 1);
    const int m = r >> 1;
    const int p = (m * s) & (kS - 1);
    float sn, cs;
    sincosf((float)p * two_pi_over_s, &sn, &cs);
    Wf[tid] = (r & 1) ? -sn : cs;
  }
  {
    // Wi[t][c]: c = 2m -> (m==0 ? 1 : 2)/S * cos ; c = 2m+1 -> -(2/S) * sin
    const int t = tid >> 6, c = tid & (kR - 1);
    const int m = c >> 1;
    const int p = (m * t) & (kS - 1);
    float sn, cs;
    sincosf((float)p * two_pi_over_s, &sn, &cs);
    const float cscale = ((m == 0) ? 1.0f : 2.0f) / (float)kS;
    Wi[tid] = (c & 1) ? (-2.0f / (float)kS) * sn : cscale * cs;
  }
}

// ---------------------------------------------------------------------------
// Stage 1: forward truncated DFT, K-split 4 ways. Grid (B, 4), 16 waves.
// Each block stages 64x64 x-chunks into padded LDS (b128 cooperative loads,
// next chunk prefetched into registers during compute), then each wave does
// 16 fp32 WMMAs per chunk on its 16x16 tile.
// ---------------------------------------------------------------------------
__global__ void __launch_bounds__(512)
fwd_dft_kernel(const float* __restrict__ x, const float* __restrict__ Wf,
               float* __restrict__ Xp) {
  __shared__ float Ls[kChunk * kPad];          // 18 KB

  const int b     = blockIdx.x;
  const int split = blockIdx.y;
  const int kbase = split * kKPer;

  const int tid  = threadIdx.x;
  const int wave = tid >> 5;
  const int lane = tid & 31;
  const int l    = lane & 15;          // M-row / N-col within tile
  const int kh   = lane >> 4;          // K-half selector (0: K=0,1  1: K=2,3)
  const int R0   = (wave >> 2) * 16;   // DFT-row tile
  const int C0   = (wave & 3) * 16;    // Cin tile

  const float* xb   = x + (size_t)b * kS * kCin + (size_t)kbase * kCin;
  const float* Arow = Wf + (size_t)(R0 + l) * kS + kbase + 2 * kh;

  // cooperative-copy index: thread -> (row, 16B column) of the 64x64 chunk
  const int crow = tid >> 4;           // 0..31 (and +32 on second op)
  const int ccol = (tid & 15) * 4;     // float column
  float* dst0 = Ls + crow * kPad + ccol;
  float* dst1 = Ls + (crow + 32) * kPad + ccol;

  // preload chunk 0
  float4 r0 = ((const float4*)xb)[tid];
  float4 r1 = ((const float4*)xb)[tid + 512];
  *(float4*)dst0 = r0;
  *(float4*)dst1 = r1;

  v8f acc = {};
  for (int ch = 0; ch < kKPer; ch += kChunk) {
    __syncthreads();                                   // chunk in LDS ready
    const bool more = (ch + kChunk) < kKPer;
    if (more) {                                        // prefetch next chunk
      const float4* src = (const float4*)(xb + (ch + kChunk) * kCin);
      r0 = src[tid];
      r1 = src[tid + 512];
    }
#pragma unroll
    for (int kk = 0; kk < kChunk; kk += 4) {
      v2f a = *(const v2f*)(Arow + ch + kk);           // K = kk+2kh, +1
      const int lrow = (kk + 2 * kh) * kPad + C0 + l;
      v2f bb;
      bb.x = Ls[lrow];
      bb.y = Ls[lrow + kPad];
      acc = wmma_f32(a, bb, acc);
    }
    __syncthreads();                                   // everyone done reading
    if (more) {
      *(float4*)dst0 = r0;
      *(float4*)dst1 = r1;
    }
  }

  float* Xb = Xp + (size_t)split * kXP + (size_t)b * kR * kCin;
#pragma unroll
  for (int v = 0; v < 8; ++v)
    Xb[(R0 + v + 8 * kh) * kCin + C0 + l] = acc[v];
}

// ---------------------------------------------------------------------------
// Stage 2: per-mode channel mix with fused 4-way partial reduction.
// One block per mode (32 blocks, 8 waves). A-rows span (b, re/im) pairs:
// q = 2*b + ri, 128 rows -> 8 row-tiles. Each wave: 1 row-tile x 4 Cout
// tiles, K = Cin = 64. A fragment = fixed-order sum of the 4 K-split
// partials (deterministic).
// ---------------------------------------------------------------------------
__global__ void __launch_bounds__(256)
mode_mix_kernel(const float* __restrict__ Xp, const float* __restrict__ Kmat,
                float* __restrict__ F) {
  const int m    = blockIdx.x;
  const int wave = threadIdx.x >> 5;   // row-tile 0..7
  const int lane = threadIdx.x & 31;
  const int l    = lane & 15;
  const int kh   = lane >> 4;

  const int q = wave * 16 + l;         // global A-row: b = q>>1, ri = q&1
  const float* Abase = Xp + (size_t)(q >> 1) * kR * kCin +
                       (2 * m + (q & 1)) * kCin + 2 * kh;
  const float* Kb = Kmat + (size_t)m * kCin * kCout + 2 * kh * kCout + l;

  v8f acc[4] = {{}, {}, {}, {}};
#pragma unroll
  for (int k0 = 0; k0 < kCin; k0 += 4) {
    v2f a0 = *(const v2f*)(Abase + k0);
    v2f a1 = *(const v2f*)(Abase + k0 + kXP);
    v2f a2 = *(const v2f*)(Abase + k0 + 2 * kXP);
    v2f a3 = *(const v2f*)(Abase + k0 + 3 * kXP);
    v2f a  = (a0 + a1) + (a2 + a3);
#pragma unroll
    for (int nc = 0; nc < 4; ++nc) {
      v2f bb;
      bb.x = Kb[k0 * kCout + nc * 16];
      bb.y = Kb[k0 * kCout + kCout + nc * 16];
      acc[nc] = wmma_f32(a, bb, acc[nc]);
    }
  }

#pragma unroll
  for (int nc = 0; nc < 4; ++nc)
#pragma unroll
    for (int v = 0; v < 8; ++v) {
      const int qq = wave * 16 + v + 8 * kh;  // output A-row
      F[(size_t)(qq >> 1) * kR * kCout + (2 * m + (qq & 1)) * kCout +
        nc * 16 + l] = acc[nc][v];
    }
}

// ---------------------------------------------------------------------------
// Stage 3: inverse DFT. Grid (B, 16) x 16 waves; each wave owns a 16-row
// time tile and all 4 Cout tiles (A fragment reused across 4 accumulators).
// F_b (16 KB) staged in LDS once per block.
// ---------------------------------------------------------------------------
__global__ void __launch_bounds__(512)
inv_dft_kernel(const float* __restrict__ Wi, const float* __restrict__ F,
               float* __restrict__ out) {
  __shared__ float Fs[kR * kCout];     // 4096 floats = 16 KB

  const int b = blockIdx.x;
  const float* Fb = F + (size_t)b * kR * kCout;
  for (int i = threadIdx.x; i < (kR * kCout) / 4; i += blockDim.x)
    ((float4*)Fs)[i] = ((const float4*)Fb)[i];
  __syncthreads();

  const int wave = threadIdx.x >> 5;
  const int lane = threadIdx.x & 31;
  const int l    = lane & 15;
  const int kh   = lane >> 4;
  const int T0   = (blockIdx.y * 16 + wave) * 16;   // time-row tile start

  const float* Abase = Wi + (size_t)(T0 + l) * kR + 2 * kh;

  v8f acc[4] = {{}, {}, {}, {}};
#pragma unroll
  for (int k0 = 0; k0 < kR; k0 += 4) {
    v2f a = *(const v2f*)(Abase + k0);
    const int krow = (k0 + 2 * kh) * kCout;
#pragma unroll
    for (int nc = 0; nc < 4; ++nc) {
      v2f bb;
      bb.x = Fs[krow + nc * 16 + l];
      bb.y = Fs[krow + kCout + nc * 16 + l];
      acc[nc] = wmma_f32(a, bb, acc[nc]);
    }
  }

  float* ob = out + (size_t)b * kS * kCout;
#pragma unroll
  for (int nc = 0; nc < 4; ++nc)
#pragma unroll
    for (int v = 0; v < 8; ++v)
      ob[(size_t)(T0 + v + 8 * kh) * kCout + nc * 16 + l] = acc[nc][v];
}

// ---------------------------------------------------------------------------
extern "C" void kernel_launch(void* const* d_in, const int* in_sizes, int n_in,
                              void* d_out, int out_size, void* d_ws,
                              size_t ws_size, hipStream_t stream) {
  const float* x    = (const float*)d_in[0];   // [64, 4096, 64] f32
  const float* kmat = (const float*)d_in[1];   // [32, 64, 64]  f32
  float* out        = (float*)d_out;           // [64, 4096, 64] f32

  char* ws = (char*)d_ws;                           // 7 MiB used
  float* Wf = (float*)(ws);                         // 64 x 4096       (1 MiB)
  float* Wi = (float*)(ws + (((size_t)1) << 20));   // 4096 x 64       (1 MiB)
  float* Xp = (float*)(ws + (((size_t)2) << 20));   // 4 x 64x64x64    (4 MiB)
  float* F  = (float*)(ws + (((size_t)6) << 20));   // 64 x 64 x 64    (1 MiB)

  twiddle_kernel<<<1024, 256, 0, stream>>>(Wf, Wi);
  fwd_dft_kernel<<<dim3(kB, kSplit), 512, 0, stream>>>(x, Wf, Xp);
  mode_mix_kernel<<<kModes, 256, 0, stream>>>(Xp, kmat, F);
  inv_dft_kernel<<<dim3(kB, 16), 512, 0, stream>>>(Wi, F, out);
}